// Pool_layer_13554916786445
// MI455X (gfx1250) — compile-verified
//
#include <hip/hip_runtime.h>
#include <float.h>

#define BS   16
#define NV   2048
#define NC   512
#define POOL 512
#define KNBR 16

typedef float v2f __attribute__((ext_vector_type(2)));
typedef float v4f __attribute__((ext_vector_type(4)));
typedef float v8f __attribute__((ext_vector_type(8)));

// One block = one (batch, 16-sample tile). 256 threads = 8 waves (wave32).
// Phase 1: 16x2048 distance tile via V_WMMA_F32_16X16X4_F32 -> LDS.
// Phase 2: per-row iterative wave argmin (17x, float4 LDS scans) -> 16 nbr idx.
// Phase 3: gather 16 feature rows, channel-wise max, store pooled row.
__global__ __launch_bounds__(256) void pool_main_kernel(
    const float* __restrict__ vtx,   // [BS][NV][3]
    const float* __restrict__ feat,  // [BS][NV][NC]
    const int*   __restrict__ sidx,  // [POOL]
    float*       __restrict__ out_feat) // [BS][POOL][NC] (d_out + BS*POOL*3)
{
    __shared__ float sdist[16 * NV];      // 128 KB distance tile
    __shared__ int   snbr[16 * KNBR];     // neighbor indices per row

    const int tile = blockIdx.x;          // 0..31  (POOL/16 sample tiles)
    const int b    = blockIdx.y;          // 0..15  batch
    const int tid  = threadIdx.x;
    const int lane = tid & 31;
    const int wave = tid >> 5;
    const int half = lane & 15;           // row/col within 16
    const bool hi  = lane >= 16;

    // ---- A matrix: 16 sampled vertices (f32 16x4, K=3 padded) ----
    const int sid = sidx[tile * 16 + half];
    const float ax = vtx[((size_t)b * NV + sid) * 3 + 0];
    const float ay = vtx[((size_t)b * NV + sid) * 3 + 1];
    const float az = vtx[((size_t)b * NV + sid) * 3 + 2];
    const float qs_l = ax * ax + ay * ay + az * az;   // |v_s|^2 for sample 'half'

    v2f A;
    A.x = hi ? az  : ax;   // VGPR0: K=0 (lanes 0-15) / K=2 (lanes 16-31)
    A.y = hi ? 0.f : ay;   // VGPR1: K=1 (lanes 0-15) / K=3 (lanes 16-31)

    // q_s for the row each D-VGPR r maps to (lo half: M=r, hi half: M=r+8)
    float qrow[8];
    #pragma unroll
    for (int r = 0; r < 8; ++r)
        qrow[r] = __shfl(qs_l, hi ? (r + 8) : r, 32);

    // ---- Phase 1: distances via WMMA, 16 column-tiles per wave ----
    #pragma unroll 1
    for (int t = 0; t < 16; ++t) {
        const int m = wave * 256 + t * 16 + half;     // candidate column
        const float cx = vtx[((size_t)b * NV + m) * 3 + 0];
        const float cy = vtx[((size_t)b * NV + m) * 3 + 1];
        const float cz = vtx[((size_t)b * NV + m) * 3 + 2];
        const float qm = cx * cx + cy * cy + cz * cz;

        v2f Bm;
        Bm.x = hi ? cz  : cx;
        Bm.y = hi ? 0.f : cy;

        v8f C = {};
        C = __builtin_amdgcn_wmma_f32_16x16x4_f32(
                /*neg_a=*/false, A, /*neg_b=*/false, Bm,
                /*c_mod=*/(short)0, C, /*reuse_a=*/false, /*reuse_b=*/false);

        #pragma unroll
        for (int r = 0; r < 8; ++r) {
            const int s = hi ? (r + 8) : r;
            sdist[s * NV + m] = qrow[r] + qm - 2.0f * C[r];
        }
    }
    __syncthreads();

    // ---- Phase 2: 17x wave argmin per row (float4 scans); record nbrs ----
    #pragma unroll 1
    for (int rr = 0; rr < 2; ++rr) {
        const int row = wave + rr * 8;
        float* drow = &sdist[row * NV];
        const v4f* drow4 = (const v4f*)drow;
        #pragma unroll 1
        for (int it = 0; it < KNBR + 1; ++it) {
            float vmin = FLT_MAX;
            int   imin = NV;
            #pragma unroll
            for (int k = 0; k < 16; ++k) {          // 16 x ds_load_b128
                const int j4 = lane + 32 * k;
                const v4f d4 = drow4[j4];
                const int jb = j4 * 4;
                if (d4.x < vmin) { vmin = d4.x; imin = jb + 0; }
                if (d4.y < vmin) { vmin = d4.y; imin = jb + 1; }
                if (d4.z < vmin) { vmin = d4.z; imin = jb + 2; }
                if (d4.w < vmin) { vmin = d4.w; imin = jb + 3; }
            }
            #pragma unroll
            for (int off = 16; off >= 1; off >>= 1) {
                const float ov = __shfl_xor(vmin, off, 32);
                const int   oi = __shfl_xor(imin, off, 32);
                if (ov < vmin || (ov == vmin && oi < imin)) { vmin = ov; imin = oi; }
            }
            if (lane == 0) {
                drow[imin] = FLT_MAX;                       // invalidate winner
                if (it > 0) snbr[row * KNBR + (it - 1)] = imin;
            }
        }
    }
    __syncthreads();

    // ---- Phase 3: gather 16 neighbor feature rows, channel-wise max ----
    #pragma unroll 1
    for (int rr = 0; rr < 2; ++rr) {
        const int row = wave + rr * 8;
        v4f acc[4];
        #pragma unroll
        for (int k = 0; k < 4; ++k)
            acc[k] = (v4f){-FLT_MAX, -FLT_MAX, -FLT_MAX, -FLT_MAX};

        #pragma unroll 1
        for (int n = 0; n < KNBR; ++n) {
            const int nb = snbr[row * KNBR + n];
            const v4f* fp = (const v4f*)(feat + ((size_t)b * NV + nb) * NC);
            if (n + 1 < KNBR) {
                const int nb2 = snbr[row * KNBR + n + 1];
                __builtin_prefetch(feat + ((size_t)b * NV + nb2) * NC, 0, 0);
            }
            #pragma unroll
            for (int k = 0; k < 4; ++k) {
                const v4f f = fp[lane + 32 * k];
                acc[k].x = fmaxf(acc[k].x, f.x);
                acc[k].y = fmaxf(acc[k].y, f.y);
                acc[k].z = fmaxf(acc[k].z, f.z);
                acc[k].w = fmaxf(acc[k].w, f.w);
            }
        }
        v4f* op = (v4f*)(out_feat + ((size_t)b * POOL + tile * 16 + row) * NC);
        #pragma unroll
        for (int k = 0; k < 4; ++k)
            op[lane + 32 * k] = acc[k];
    }
}

// vertices_pool: out[b][s][d] = vtx[b][sidx[s]][d]  (BS*POOL*3 elements)
__global__ __launch_bounds__(256) void vert_pool_kernel(
    const float* __restrict__ vtx, const int* __restrict__ sidx,
    float* __restrict__ out)
{
    const int i = blockIdx.x * blockDim.x + threadIdx.x;
    if (i >= BS * POOL * 3) return;
    const int d = i % 3;
    const int s = (i / 3) % POOL;
    const int b = i / (3 * POOL);
    out[i] = vtx[((size_t)b * NV + sidx[s]) * 3 + d];
}

extern "C" void kernel_launch(void* const* d_in, const int* in_sizes, int n_in,
                              void* d_out, int out_size, void* d_ws, size_t ws_size,
                              hipStream_t stream) {
    (void)in_sizes; (void)n_in; (void)out_size; (void)d_ws; (void)ws_size;
    const float* vtx  = (const float*)d_in[0];   // 16*2048*3 f32
    const float* feat = (const float*)d_in[1];   // 16*2048*512 f32
    const int*   sidx = (const int*)d_in[2];     // 512 i32

    float* out_vert = (float*)d_out;                         // 16*512*3
    float* out_feat = (float*)d_out + (size_t)BS * POOL * 3; // 16*512*512

    dim3 grid_v((BS * POOL * 3 + 255) / 256);
    vert_pool_kernel<<<grid_v, 256, 0, stream>>>(vtx, sidx, out_vert);

    dim3 grid_m(POOL / 16, BS);   // 32 sample tiles x 16 batches
    pool_main_kernel<<<grid_m, 256, 0, stream>>>(vtx, feat, sidx, out_feat);
}